// LGEB_8770323219157
// MI455X (gfx1250) — compile-verified
//
#include <hip/hip_runtime.h>
#include <hip/hip_bf16.h>
#include <math.h>

typedef float v2f __attribute__((ext_vector_type(2)));
typedef float v8f __attribute__((ext_vector_type(8)));

#define NNODE 50000
#define NEDGE 800000
#define NHID  64
#define ETILES 50000   // NEDGE/16
#define NTILES 3125    // NNODE/16

// ---------------------------------------------------------------------------
// V_WMMA_F32_16X16X4_F32: D(16x16 f32) = A(16x4 f32) * B(4x16 f32) + C
// A frag: lane L<16 -> row L, {K=k,k+1}; lane 16+L -> row L, {K=k+2,k+3}
// B frag: lane n<16 -> col n, {K=k,k+1}; lane 16+n -> col n, {K=k+2,k+3}
// C/D:    VGPR r, lanes 0-15: (M=r,   N=lane); lanes 16-31: (M=r+8, N=lane-16)
// ---------------------------------------------------------------------------
__device__ __forceinline__ v8f wmma_f32_k4(v2f a, v2f b, v8f c) {
  return __builtin_amdgcn_wmma_f32_16x16x4_f32(false, a, false, b, (short)0, c,
                                               false, false);
}

__device__ __forceinline__ float psi_f(float p) {
  return copysignf(logf(fabsf(p) + 1.0f), p);
}

// ---------------------------------------------------------------------------
__global__ void k_zero(float* __restrict__ p, size_t n) {
  size_t i = (size_t)blockIdx.x * blockDim.x + threadIdx.x;
  size_t st = (size_t)gridDim.x * blockDim.x;
  for (; i < n; i += st) p[i] = 0.0f;
}

// ---------------------------------------------------------------------------
// Edge pass 1: z = feat @ W_e1  (feat = [h_i | h_j | psi(norm) | psi(dot)],
// K = 130, padded to 132). Also per-block BN partial sums (sum, sumsq).
// Grid: 12500 blocks x 128 threads (4 waves, 1 tile of 16 edges each).
// ---------------------------------------------------------------------------
__global__ void __launch_bounds__(128) k_edge_gemm1(
    const float* __restrict__ h, const float* __restrict__ x,
    const int* __restrict__ ei, const int* __restrict__ ej,
    const float* __restrict__ We1,          // [130][64]
    float* __restrict__ z,                  // [E][64] (m-region of d_out)
    float* __restrict__ part)               // [gridDim][128]
{
  __shared__ float s_stats[128];
  const int tid  = threadIdx.x;
  const int wave = tid >> 5;
  const int lane = tid & 31;
  const int n    = lane & 15;
  const int g    = lane >> 4;
  const int koff = g * 2;
  const int tile = blockIdx.x * 4 + wave;   // always < ETILES
  const int row0 = tile * 16;
  const int arow = row0 + n;

  const int e0 = ei[arow];
  const int e1 = ej[arow];
  const float* pi = h + (size_t)e0 * NHID;
  const float* pj = h + (size_t)e1 * NHID;

  const float4* x4 = (const float4*)x;
  float4 xi = x4[e0], xj = x4[e1];
  float d0 = xi.x - xj.x, d1 = xi.y - xj.y, d2 = xi.z - xj.z, d3 = xi.w - xj.w;
  float nrm = psi_f(d0 * d0 - d1 * d1 - d2 * d2 - d3 * d3);
  float dot = psi_f(xi.x * xj.x - xi.y * xj.y - xi.z * xj.z - xi.w * xj.w);

  v8f acc[4] = {};

#pragma unroll 4
  for (int s = 0; s < 16; ++s) {            // K = 0..63 from h[ei]
    int kk = 4 * s + koff;
    v2f a = *(const v2f*)(pi + kk);
#pragma unroll
    for (int nt = 0; nt < 4; ++nt) {
      int col = nt * 16 + n;
      v2f b;
      b.x = We1[kk * NHID + col];
      b.y = We1[(kk + 1) * NHID + col];
      acc[nt] = wmma_f32_k4(a, b, acc[nt]);
    }
  }
#pragma unroll 4
  for (int s = 16; s < 32; ++s) {           // K = 64..127 from h[ej]
    int kk = 4 * s + koff;                  // W row index
    v2f a = *(const v2f*)(pj + (kk - 64));
#pragma unroll
    for (int nt = 0; nt < 4; ++nt) {
      int col = nt * 16 + n;
      v2f b;
      b.x = We1[kk * NHID + col];
      b.y = We1[(kk + 1) * NHID + col];
      acc[nt] = wmma_f32_k4(a, b, acc[nt]);
    }
  }
  {                                         // K = 128..131 (130,131 zero-pad)
    v2f a;
    a.x = g ? 0.0f : nrm;
    a.y = g ? 0.0f : dot;
#pragma unroll
    for (int nt = 0; nt < 4; ++nt) {
      int col = nt * 16 + n;
      v2f b;
      b.x = g ? 0.0f : We1[128 * NHID + col];
      b.y = g ? 0.0f : We1[129 * NHID + col];
      acc[nt] = wmma_f32_k4(a, b, acc[nt]);
    }
  }

  if (tid < 128) s_stats[tid] = 0.0f;
  __syncthreads();
#pragma unroll
  for (int nt = 0; nt < 4; ++nt) {
    int col = nt * 16 + n;
    float sum = 0.0f, sq = 0.0f;
#pragma unroll
    for (int r = 0; r < 8; ++r) {
      float v = acc[nt][r];
      z[(size_t)(row0 + r + 8 * g) * NHID + col] = v;
      sum += v;
      sq += v * v;
    }
    atomicAdd(&s_stats[col], sum);
    atomicAdd(&s_stats[64 + col], sq);
  }
  __syncthreads();
  if (tid < 128) part[(size_t)blockIdx.x * 128 + tid] = s_stats[tid];
}

// ---------------------------------------------------------------------------
// Reduce per-block partials [nparts][128] -> sums[128]. Grid: 128 x 256.
// ---------------------------------------------------------------------------
__global__ void k_reduce_stats(const float* __restrict__ part, int nparts,
                               float* __restrict__ sums) {
  __shared__ float red[256];
  const int c = blockIdx.x;
  float s = 0.0f;
  for (int i = threadIdx.x; i < nparts; i += blockDim.x)
    s += part[(size_t)i * 128 + c];
  red[threadIdx.x] = s;
  __syncthreads();
  for (int off = 128; off > 0; off >>= 1) {
    if (threadIdx.x < off) red[threadIdx.x] += red[threadIdx.x + off];
    __syncthreads();
  }
  if (threadIdx.x == 0) sums[c] = red[0];
}

// sums[0:64]=sum, sums[64:128]=sumsq; biased var + eps=1e-5 (torch BN train).
__global__ void k_bn_finalize(const float* __restrict__ sums, float count,
                              const float* __restrict__ gamma,
                              const float* __restrict__ beta,
                              float* __restrict__ scale,
                              float* __restrict__ shift) {
  int c = threadIdx.x;                      // 64 threads
  float mu = sums[c] / count;
  float var = sums[64 + c] / count - mu * mu;
  float rstd = rsqrtf(fmaxf(var, 0.0f) + 1e-5f);
  float sc = gamma[c] * rstd;
  scale[c] = sc;
  shift[c] = beta[c] - mu * sc;
}

// ---------------------------------------------------------------------------
// Edge pass 2: out = relu(relu(BN(z)) @ We2 + be2); gate = sigmoid(out@Wm+bm);
// m = out*gate -> d_out & atomic agg_h; hid = relu(m@Wx1+bx1);
// phx = hid@Wx2; trans = clip(xd*phx) -> atomic agg_x, cnt.
// ---------------------------------------------------------------------------
__global__ void __launch_bounds__(128) k_edge_pass2(
    float* __restrict__ z_m,                // in: z, out: m  [E][64]
    const float* __restrict__ We2, const float* __restrict__ be2,
    const float* __restrict__ Wm, const float* __restrict__ bm,
    const float* __restrict__ Wx1, const float* __restrict__ bx1,
    const float* __restrict__ Wx2,
    const float* __restrict__ x,
    const int* __restrict__ ei, const int* __restrict__ ej,
    const float* __restrict__ esc, const float* __restrict__ esh,
    float* __restrict__ aggh, float* __restrict__ aggx,
    float* __restrict__ cnt)
{
  __shared__ float s_tile[4][16][68];       // stride 68 -> conflict-free
  __shared__ float s_vec[4][16];
  const int tid  = threadIdx.x;
  const int wave = tid >> 5;
  const int lane = tid & 31;
  const int n    = lane & 15;
  const int g    = lane >> 4;
  const int koff = g * 2;
  const int tile = blockIdx.x * 4 + wave;   // always < ETILES
  const int row0 = tile * 16;

  // GEMM2 with bias preloaded in C
  v8f acc[4];
#pragma unroll
  for (int nt = 0; nt < 4; ++nt) {
    float bc = be2[nt * 16 + n];
#pragma unroll
    for (int r = 0; r < 8; ++r) acc[nt][r] = bc;
  }
  const float* zr = z_m + (size_t)(row0 + n) * NHID;
#pragma unroll 4
  for (int s = 0; s < 16; ++s) {
    int kk = 4 * s + koff;
    v2f zv = *(const v2f*)(zr + kk);
    v2f a;
    a.x = fmaxf(zv.x * esc[kk] + esh[kk], 0.0f);
    a.y = fmaxf(zv.y * esc[kk + 1] + esh[kk + 1], 0.0f);
#pragma unroll
    for (int nt = 0; nt < 4; ++nt) {
      int col = nt * 16 + n;
      v2f b;
      b.x = We2[kk * NHID + col];
      b.y = We2[(kk + 1) * NHID + col];
      acc[nt] = wmma_f32_k4(a, b, acc[nt]);
    }
  }
  // ReLU, stash "out" tile in LDS for the row-wise gate reduction
#pragma unroll
  for (int nt = 0; nt < 4; ++nt) {
    int col = nt * 16 + n;
#pragma unroll
    for (int r = 0; r < 8; ++r) {
      float v = fmaxf(acc[nt][r], 0.0f);
      acc[nt][r] = v;
      s_tile[wave][r + 8 * g][col] = v;
    }
  }
  __syncthreads();
  if (lane < 16) {
    float t = bm[0];
#pragma unroll 8
    for (int c = 0; c < NHID; ++c) t += s_tile[wave][lane][c] * Wm[c];
    s_vec[wave][lane] = 1.0f / (1.0f + expf(-t));
  }
  __syncthreads();
  float gates[8];
  int rowe[8];
#pragma unroll
  for (int r = 0; r < 8; ++r) {
    gates[r] = s_vec[wave][r + 8 * g];
    rowe[r] = ei[row0 + r + 8 * g];
  }
  // m = out * gate: write to d_out, scatter into agg_h, stash in LDS
#pragma unroll
  for (int nt = 0; nt < 4; ++nt) {
    int col = nt * 16 + n;
#pragma unroll
    for (int r = 0; r < 8; ++r) {
      float mv = acc[nt][r] * gates[r];
      z_m[(size_t)(row0 + r + 8 * g) * NHID + col] = mv;
      s_tile[wave][r + 8 * g][col] = mv;
      atomicAdd(&aggh[(size_t)rowe[r] * NHID + col], mv);
    }
  }
  __syncthreads();
  // GEMM3: hid = relu(m @ Wx1 + bx1), A-fragments from LDS m tile
  v8f hacc[4];
#pragma unroll
  for (int nt = 0; nt < 4; ++nt) {
    float bc = bx1[nt * 16 + n];
#pragma unroll
    for (int r = 0; r < 8; ++r) hacc[nt][r] = bc;
  }
#pragma unroll 4
  for (int s = 0; s < 16; ++s) {
    int kk = 4 * s + koff;
    v2f a = *(const v2f*)&s_tile[wave][n][kk];
#pragma unroll
    for (int nt = 0; nt < 4; ++nt) {
      int col = nt * 16 + n;
      v2f b;
      b.x = Wx1[kk * NHID + col];
      b.y = Wx1[(kk + 1) * NHID + col];
      hacc[nt] = wmma_f32_k4(a, b, hacc[nt]);
    }
  }
  __syncthreads();
#pragma unroll
  for (int nt = 0; nt < 4; ++nt)
#pragma unroll
    for (int r = 0; r < 8; ++r)
      s_tile[wave][r + 8 * g][nt * 16 + n] = fmaxf(hacc[nt][r], 0.0f);
  __syncthreads();
  if (lane < 16) {
    int row = row0 + lane;
    float ph = 0.0f;
#pragma unroll 8
    for (int c = 0; c < NHID; ++c) ph += s_tile[wave][lane][c] * Wx2[c];
    int e0 = ei[row], e1 = ej[row];
    const float4* x4 = (const float4*)x;
    float4 xi = x4[e0], xj = x4[e1];
    float t0 = fminf(fmaxf((xi.x - xj.x) * ph, -100.0f), 100.0f);
    float t1 = fminf(fmaxf((xi.y - xj.y) * ph, -100.0f), 100.0f);
    float t2 = fminf(fmaxf((xi.z - xj.z) * ph, -100.0f), 100.0f);
    float t3 = fminf(fmaxf((xi.w - xj.w) * ph, -100.0f), 100.0f);
    atomicAdd(&aggx[(size_t)e0 * 4 + 0], t0);
    atomicAdd(&aggx[(size_t)e0 * 4 + 1], t1);
    atomicAdd(&aggx[(size_t)e0 * 4 + 2], t2);
    atomicAdd(&aggx[(size_t)e0 * 4 + 3], t3);
    atomicAdd(&cnt[e0], 1.0f);
  }
}

// ---------------------------------------------------------------------------
// Node pass 1: zh = [h | agg_h | node_attr] @ Wh1 + bh1 (K=136) + BN partials
// ---------------------------------------------------------------------------
__global__ void __launch_bounds__(128) k_node_gemm1(
    const float* __restrict__ h, const float* __restrict__ aggh,
    const float* __restrict__ nattr,
    const float* __restrict__ Wh1, const float* __restrict__ bh1,
    float* __restrict__ zh, float* __restrict__ part, int ntiles)
{
  __shared__ float s_stats[128];
  const int tid  = threadIdx.x;
  const int wave = tid >> 5;
  const int lane = tid & 31;
  const int n    = lane & 15;
  const int g    = lane >> 4;
  const int koff = g * 2;
  const int tile = blockIdx.x * 4 + wave;
  const bool active = tile < ntiles;        // wave-uniform
  const int tc = active ? tile : (ntiles - 1);
  const int row0 = tc * 16;
  const int arow = row0 + n;

  const float* ph = h + (size_t)arow * NHID;
  const float* pa = aggh + (size_t)arow * NHID;
  const float* pt = nattr + (size_t)arow * 8;

  v8f acc[4];
#pragma unroll
  for (int nt = 0; nt < 4; ++nt) {
    float bc = bh1[nt * 16 + n];
#pragma unroll
    for (int r = 0; r < 8; ++r) acc[nt][r] = bc;
  }
#pragma unroll 4
  for (int s = 0; s < 16; ++s) {            // K = 0..63 from h
    int kk = 4 * s + koff;
    v2f a = *(const v2f*)(ph + kk);
#pragma unroll
    for (int nt = 0; nt < 4; ++nt) {
      int col = nt * 16 + n;
      v2f b;
      b.x = Wh1[kk * NHID + col];
      b.y = Wh1[(kk + 1) * NHID + col];
      acc[nt] = wmma_f32_k4(a, b, acc[nt]);
    }
  }
#pragma unroll 4
  for (int s = 16; s < 32; ++s) {           // K = 64..127 from agg_h
    int kk = 4 * s + koff;
    v2f a = *(const v2f*)(pa + (kk - 64));
#pragma unroll
    for (int nt = 0; nt < 4; ++nt) {
      int col = nt * 16 + n;
      v2f b;
      b.x = Wh1[kk * NHID + col];
      b.y = Wh1[(kk + 1) * NHID + col];
      acc[nt] = wmma_f32_k4(a, b, acc[nt]);
    }
  }
#pragma unroll
  for (int s = 32; s < 34; ++s) {           // K = 128..135 from node_attr
    int kk = 4 * s + koff;
    v2f a = *(const v2f*)(pt + (kk - 128));
#pragma unroll
    for (int nt = 0; nt < 4; ++nt) {
      int col = nt * 16 + n;
      v2f b;
      b.x = Wh1[kk * NHID + col];
      b.y = Wh1[(kk + 1) * NHID + col];
      acc[nt] = wmma_f32_k4(a, b, acc[nt]);
    }
  }

  if (tid < 128) s_stats[tid] = 0.0f;
  __syncthreads();
  if (active) {
#pragma unroll
    for (int nt = 0; nt < 4; ++nt) {
      int col = nt * 16 + n;
      float sum = 0.0f, sq = 0.0f;
#pragma unroll
      for (int r = 0; r < 8; ++r) {
        float v = acc[nt][r];
        zh[(size_t)(row0 + r + 8 * g) * NHID + col] = v;
        sum += v;
        sq += v * v;
      }
      atomicAdd(&s_stats[col], sum);
      atomicAdd(&s_stats[64 + col], sq);
    }
  }
  __syncthreads();
  if (tid < 128) part[(size_t)blockIdx.x * 128 + tid] = s_stats[tid];
}

// ---------------------------------------------------------------------------
// Node pass 2: h_new = h + relu(BN(zh)) @ Wh2 + bh2; x_new = x + agg_x/cnt
// ---------------------------------------------------------------------------
__global__ void __launch_bounds__(128) k_node_pass2(
    const float* __restrict__ zh,
    const float* __restrict__ nsc, const float* __restrict__ nsh,
    const float* __restrict__ Wh2, const float* __restrict__ bh2,
    const float* __restrict__ h, const float* __restrict__ x,
    const float* __restrict__ aggx, const float* __restrict__ cnt,
    float* __restrict__ hout, float* __restrict__ xout, int ntiles)
{
  const int tid  = threadIdx.x;
  const int wave = tid >> 5;
  const int lane = tid & 31;
  const int n    = lane & 15;
  const int g    = lane >> 4;
  const int koff = g * 2;
  const int tile = blockIdx.x * 4 + wave;
  const bool active = tile < ntiles;
  const int tc = active ? tile : (ntiles - 1);
  const int row0 = tc * 16;
  const int arow = row0 + n;

  v8f acc[4];
#pragma unroll
  for (int nt = 0; nt < 4; ++nt) {
    int col = nt * 16 + n;
    float bc = bh2[col];
#pragma unroll
    for (int r = 0; r < 8; ++r)             // residual: C = bias + h
      acc[nt][r] = bc + h[(size_t)(row0 + r + 8 * g) * NHID + col];
  }
  const float* zr = zh + (size_t)arow * NHID;
#pragma unroll 4
  for (int s = 0; s < 16; ++s) {
    int kk = 4 * s + koff;
    v2f zv = *(const v2f*)(zr + kk);
    v2f a;
    a.x = fmaxf(zv.x * nsc[kk] + nsh[kk], 0.0f);
    a.y = fmaxf(zv.y * nsc[kk + 1] + nsh[kk + 1], 0.0f);
#pragma unroll
    for (int nt = 0; nt < 4; ++nt) {
      int col = nt * 16 + n;
      v2f b;
      b.x = Wh2[kk * NHID + col];
      b.y = Wh2[(kk + 1) * NHID + col];
      acc[nt] = wmma_f32_k4(a, b, acc[nt]);
    }
  }
  if (active) {
#pragma unroll
    for (int nt = 0; nt < 4; ++nt) {
      int col = nt * 16 + n;
#pragma unroll
      for (int r = 0; r < 8; ++r)
        hout[(size_t)(row0 + r + 8 * g) * NHID + col] = acc[nt][r];
    }
    if (lane < 16) {
      int row = row0 + lane;
      float c = fmaxf(cnt[row], 1.0f);
      const float4* x4 = (const float4*)x;
      const float4* a4 = (const float4*)aggx;
      float4 xv = x4[row], av = a4[row];
      xout[(size_t)row * 4 + 0] = xv.x + av.x / c;
      xout[(size_t)row * 4 + 1] = xv.y + av.y / c;
      xout[(size_t)row * 4 + 2] = xv.z + av.z / c;
      xout[(size_t)row * 4 + 3] = xv.w + av.w / c;
    }
  }
}

// ---------------------------------------------------------------------------
extern "C" void kernel_launch(void* const* d_in, const int* in_sizes, int n_in,
                              void* d_out, int out_size, void* d_ws,
                              size_t ws_size, hipStream_t stream) {
  const float* h    = (const float*)d_in[0];
  const float* x    = (const float*)d_in[1];
  const int*   ei   = (const int*)d_in[2];
  const int*   ej   = (const int*)d_in[3];
  const float* natt = (const float*)d_in[4];
  const float* We1  = (const float*)d_in[5];
  const float* ge   = (const float*)d_in[6];
  const float* be   = (const float*)d_in[7];
  const float* We2  = (const float*)d_in[8];
  const float* be2  = (const float*)d_in[9];
  const float* Wm   = (const float*)d_in[10];
  const float* bm   = (const float*)d_in[11];
  const float* Wx1  = (const float*)d_in[12];
  const float* bx1  = (const float*)d_in[13];
  const float* Wx2  = (const float*)d_in[14];
  const float* Wh1  = (const float*)d_in[15];
  const float* bh1  = (const float*)d_in[16];
  const float* gh   = (const float*)d_in[17];
  const float* bh   = (const float*)d_in[18];
  const float* Wh2  = (const float*)d_in[19];
  const float* bh2  = (const float*)d_in[20];

  float* out  = (float*)d_out;
  float* hout = out;                                  // [N][64]
  float* xout = out + (size_t)NNODE * 64;             // [N][4]
  float* mbuf = out + (size_t)NNODE * 68;             // [E][64] (z scratch -> m)

  float* ws    = (float*)d_ws;
  float* cnt   = ws;                                  // [N]
  float* aggx  = ws + NNODE;                          // [N][4]
  float* aggh  = ws + (size_t)5 * NNODE;              // [N][64]
  float* zh    = ws + (size_t)69 * NNODE;             // [N][64]
  float* esc   = ws + (size_t)133 * NNODE;            // [64]
  float* esh   = esc + 64;
  float* nsc   = esh + 64;
  float* nsh   = nsc + 64;
  float* esums = nsh + 64;                            // [128]
  float* nsums = esums + 128;                         // [128]
  float* epart = nsums + 128;                         // [12500][128]
  float* npart = epart + (size_t)12500 * 128;         // [782][128]

  const int EBLK = ETILES / 4;                        // 12500
  const int NBLK = (NTILES + 3) / 4;                  // 782

  k_zero<<<2048, 256, 0, stream>>>(ws, (size_t)69 * NNODE);
  k_edge_gemm1<<<EBLK, 128, 0, stream>>>(h, x, ei, ej, We1, mbuf, epart);
  k_reduce_stats<<<128, 256, 0, stream>>>(epart, EBLK, esums);
  k_bn_finalize<<<1, 64, 0, stream>>>(esums, (float)NEDGE, ge, be, esc, esh);
  k_edge_pass2<<<EBLK, 128, 0, stream>>>(mbuf, We2, be2, Wm, bm, Wx1, bx1, Wx2,
                                         x, ei, ej, esc, esh, aggh, aggx, cnt);
  k_node_gemm1<<<NBLK, 128, 0, stream>>>(h, aggh, natt, Wh1, bh1, zh, npart,
                                         NTILES);
  k_reduce_stats<<<128, 256, 0, stream>>>(npart, NBLK, nsums);
  k_bn_finalize<<<1, 64, 0, stream>>>(nsums, (float)NNODE, gh, bh, nsc, nsh);
  k_node_pass2<<<NBLK, 128, 0, stream>>>(zh, nsc, nsh, Wh2, bh2, h, x, aggx,
                                         cnt, hout, xout, NTILES);
}